// ComplexGraph_19791209300074
// MI455X (gfx1250) — compile-verified
//
#include <hip/hip_runtime.h>
#include <hip/hip_bf16.h>

typedef __attribute__((ext_vector_type(2))) float v2f;
typedef __attribute__((ext_vector_type(8))) float v8f;
typedef __attribute__((ext_vector_type(4))) unsigned int u32x4;
typedef __attribute__((ext_vector_type(8))) int i32x8;
typedef __attribute__((ext_vector_type(4))) int i32x4;

#define NB        64
#define NPG       512
#define NTOT      (NB * NPG)
#define LIG       128
#define TPG       32          /* 512/16 tiles per graph per dimension */
#define LTPG      8           /* ligand tiles = LIG/16 (tile-aligned!) */
#define CTX_CUT2  64.0f       /* 8^2  */
#define INTER_CUT2 100.0f     /* 10^2 */
#define GN_PER_GRAPH 1020     /* global<->same-segment edges per graph */
#define GG_PER_GRAPH 2        /* global<->global edges per graph */

// ---------------------------------------------------------------------------
// Pass 0: pack coords + squared norm into float4.
// ---------------------------------------------------------------------------
__global__ void prep_kernel(const float* __restrict__ X, float4* __restrict__ xs4) {
  int i = blockIdx.x * blockDim.x + threadIdx.x;
  if (i >= NTOT) return;
  float x = X[3 * i + 0], y = X[3 * i + 1], z = X[3 * i + 2];
  xs4[i] = make_float4(x, y, z, x * x + y * y + z * z);
}

// ---------------------------------------------------------------------------
// TDM: stage one graph's 512 float4 nodes (8 KB) into LDS via the Tensor
// Data Mover. D# group0/group1 built per CDNA5 ISA §8.3/8.4 as a 1D tile of
// 1024 x 8-byte elements; lds_addr = 0 (sx[] is the only LDS object).
// EXEC is ignored by TENSOR_LOAD_TO_LDS; it issues once per wave.
// amdgpu-toolchain (clang-23) 6-arg builtin form.
// ---------------------------------------------------------------------------
__device__ __forceinline__ void tdm_load_graph_to_lds(const float4* gsrc) {
  unsigned long long ga = (unsigned long long)(size_t)gsrc;
  u32x4 g0;
  g0[0] = 1u;                                     // count=1 (valid), user mode
  g0[1] = 0u;                                     // lds_addr = 0
  g0[2] = (unsigned)(ga & 0xFFFFFFFFu);           // global_addr[31:0]
  g0[3] = (unsigned)((ga >> 32) & 0x01FFFFFFu)    // global_addr[56:32]
        | (2u << 30);                             // type = 2 ("image")
  i32x8 g1;
  g1[0] = (3 << 16);                              // data_size=3 (8B), mask=0
  g1[1] = (int)(1024u << 16);                     // tensor_dim0 = 1024 (lo16)
  g1[2] = (int)(1u << 16);                        // dim0 hi=0 | tensor_dim1=1
  g1[3] = (int)(1024u << 16);                     // dim1 hi=0 | tile_dim0=1024
  g1[4] = 1;                                      // tile_dim1=1, tile_dim2=0
  g1[5] = 1024;                                   // tensor_dim0_stride = 1024
  g1[6] = 0;
  g1[7] = 0;
  i32x4 gz4; gz4[0] = 0; gz4[1] = 0; gz4[2] = 0; gz4[3] = 0;
  i32x8 gz8; gz8[0] = 0; gz8[1] = 0; gz8[2] = 0; gz8[3] = 0;
             gz8[4] = 0; gz8[5] = 0; gz8[6] = 0; gz8[7] = 0;
  __builtin_amdgcn_tensor_load_to_lds(g0, g1, gz4, gz4, gz8, 0);
  __builtin_amdgcn_s_wait_tensorcnt(0);
}

// ---------------------------------------------------------------------------
// Passes 1 & 3: one wave per (graph, 16-row tile).
//   dist2 tile = WMMA(A=(x,y,z,1), B=(-2x,-2y,-2z,|xj|^2), C=accn)
// where accn[i][j] = |xi|^2 comes from a loop-invariant broadcast WMMA
// (A'=(|xi|^2,0,0,0), B'=ones-row). Region-specialized by the tile-aligned
// segment split: ligand rows -> inter(+reduced) vs protein cols only;
// protein rows -> inter vs ligand cols, ctx vs protein cols.
// EMIT=0: count per-row matches. EMIT=1: write at scanned offsets, columns
// ascending -> exact jnp.nonzero row-major order.
// ---------------------------------------------------------------------------
template <int EMIT>
__global__ __launch_bounds__(32) void tiles_kernel(
    const float4* __restrict__ xs4,
    int* __restrict__ rc_ctx, int* __restrict__ rc_inter, int* __restrict__ rc_half,
    const int* __restrict__ ro_ctx, const int* __restrict__ ro_inter,
    const int* __restrict__ ro_half,
    const int* __restrict__ header, int* __restrict__ out)
{
  __shared__ float4 sx[NPG];
  const int lane = threadIdx.x;
  const int g    = blockIdx.x / TPG;
  const int rt   = blockIdx.x % TPG;
  const int base = g * NPG;
  const int l16  = lane & 15;

  tdm_load_graph_to_lds(xs4 + base);

  const int    rl = rt * 16 + l16;
  const float4 rf = sx[rl];

  // A operand (16x4 f32): lanes 0-15 hold (K0,K1)=(x,y) of row M=lane;
  // lanes 16-31 hold (K2,K3)=(z,1) of row M=lane-16.
  v2f a;
  if (lane < 16) { a[0] = rf.x; a[1] = rf.y; }
  else           { a[0] = rf.z; a[1] = 1.0f; }

  // Broadcast |x_row|^2 into all columns: accn[i][j] = n_i (loop-invariant C).
  v2f an, bn;
  if (lane < 16) { an[0] = rf.w; an[1] = 0.0f; bn[0] = 1.0f; bn[1] = 0.0f; }
  else           { an[0] = 0.0f; an[1] = 0.0f; bn[0] = 0.0f; bn[1] = 0.0f; }
  v8f zero8 = {};
  const v8f accn = __builtin_amdgcn_wmma_f32_16x16x4_f32(
      false, an, false, bn, (short)0, zero8, false, false);

  const int rowHalf = (lane >> 4) << 3;     // +0 / +8 row offset of this lane's elems

  int cnt_ctx = 0, cnt_inter = 0;
  int pos_ctx = 0, pos_inter = 0, pos_half = 0;
  int Ectx = 0, Einter = 0, Ehalf = 0, b2 = 0, b3 = 0;
  if (EMIT) {
    Ectx = header[3]; Einter = header[1]; Ehalf = header[2];
    b2 = 2 * Ectx; b3 = b2 + 2 * Einter;
    if (lane < 16) {                        // lane r owns tile row r
      pos_ctx   = ro_ctx  [base + rt * 16 + lane];
      pos_inter = ro_inter[base + rt * 16 + lane];
      pos_half  = ro_half [base + rt * 16 + lane];
    }
  }
  const int rowg = base + rt * 16 + lane;   // row global index (owner lanes 0-15)

  if (rt < LTPG) {
    // ---- ligand rows: inter edges vs protein cols; row<col always =>
    //      every edge also feeds the reduced (half) arrays.
    for (int ct = LTPG; ct < TPG; ++ct) {
      const int    cl = ct * 16 + l16;
      const float4 cf = sx[cl];
      v2f b;
      if (lane < 16) { b[0] = -2.0f * cf.x; b[1] = -2.0f * cf.y; }
      else           { b[0] = -2.0f * cf.z; b[1] = cf.w; }
      const v8f d = __builtin_amdgcn_wmma_f32_16x16x4_f32(
          false, a, false, b, (short)0, accn, false, false);
#pragma unroll
      for (int v = 0; v < 8; ++v) {
        const int  lrow = rt * 16 + v + rowHalf;
        const bool pi = (d[v] <= INTER_CUT2) && (lrow != 0) && (cl != LIG);
        const unsigned mi = (unsigned)__ballot(pi);
        if (!EMIT) {
          cnt_inter += ((lane == v)     ? __popc(mi & 0xFFFFu) : 0)
                     + ((lane == v + 8) ? __popc(mi >> 16)     : 0);
        } else if (lane == v || lane == v + 8) {
          unsigned bits = (mi >> ((lane == v) ? 0 : 16)) & 0xFFFFu;
          while (bits) {
            int c = __ffs(bits) - 1; bits &= bits - 1;
            out[b2 + pos_inter]          = rowg;
            out[b2 + Einter + pos_inter] = base + ct * 16 + c;
            ++pos_inter;
            out[b3 + pos_half]         = g;     // reduced_inter_edge_batchid
            out[b3 + Ehalf + pos_half] = base;  // reduced_inter_edge_offsets
            ++pos_half;
          }
        }
      }
    }
  } else {
    // ---- protein rows, ligand cols: inter edges (row>col, never reduced).
    for (int ct = 0; ct < LTPG; ++ct) {
      const int    cl = ct * 16 + l16;
      const float4 cf = sx[cl];
      v2f b;
      if (lane < 16) { b[0] = -2.0f * cf.x; b[1] = -2.0f * cf.y; }
      else           { b[0] = -2.0f * cf.z; b[1] = cf.w; }
      const v8f d = __builtin_amdgcn_wmma_f32_16x16x4_f32(
          false, a, false, b, (short)0, accn, false, false);
#pragma unroll
      for (int v = 0; v < 8; ++v) {
        const int  lrow = rt * 16 + v + rowHalf;
        const bool pi = (d[v] <= INTER_CUT2) && (lrow != LIG) && (cl != 0);
        const unsigned mi = (unsigned)__ballot(pi);
        if (!EMIT) {
          cnt_inter += ((lane == v)     ? __popc(mi & 0xFFFFu) : 0)
                     + ((lane == v + 8) ? __popc(mi >> 16)     : 0);
        } else if (lane == v || lane == v + 8) {
          unsigned bits = (mi >> ((lane == v) ? 0 : 16)) & 0xFFFFu;
          while (bits) {
            int c = __ffs(bits) - 1; bits &= bits - 1;
            out[b2 + pos_inter]          = rowg;
            out[b2 + Einter + pos_inter] = base + ct * 16 + c;
            ++pos_inter;
          }
        }
      }
    }
    // ---- protein rows, protein cols: ctx edges only.
    for (int ct = LTPG; ct < TPG; ++ct) {
      const int    cl = ct * 16 + l16;
      const float4 cf = sx[cl];
      v2f b;
      if (lane < 16) { b[0] = -2.0f * cf.x; b[1] = -2.0f * cf.y; }
      else           { b[0] = -2.0f * cf.z; b[1] = cf.w; }
      const v8f d = __builtin_amdgcn_wmma_f32_16x16x4_f32(
          false, a, false, b, (short)0, accn, false, false);
#pragma unroll
      for (int v = 0; v < 8; ++v) {
        const int  lrow = rt * 16 + v + rowHalf;
        const bool pc = (d[v] <= CTX_CUT2) && (lrow != LIG) && (cl != LIG) &&
                        (lrow != cl);
        const unsigned mc = (unsigned)__ballot(pc);
        if (!EMIT) {
          cnt_ctx += ((lane == v)     ? __popc(mc & 0xFFFFu) : 0)
                   + ((lane == v + 8) ? __popc(mc >> 16)     : 0);
        } else if (lane == v || lane == v + 8) {
          unsigned bits = (mc >> ((lane == v) ? 0 : 16)) & 0xFFFFu;
          while (bits) {
            int c = __ffs(bits) - 1; bits &= bits - 1;
            out[pos_ctx]        = rowg;
            out[Ectx + pos_ctx] = base + ct * 16 + c;
            ++pos_ctx;
          }
        }
      }
    }
  }

  if (!EMIT && lane < 16) {
    const int r = base + rt * 16 + lane;
    rc_ctx  [r] = cnt_ctx;
    rc_inter[r] = cnt_inter;
    rc_half [r] = (rt < LTPG) ? cnt_inter : 0;  // ligand rows: all inter are half
  }
}

// ---------------------------------------------------------------------------
// Pass 2: single-block exclusive scan of the 32768 per-row counts (x3),
// plus output-section totals in header.
// ---------------------------------------------------------------------------
__global__ __launch_bounds__(1024) void scan_kernel(
    const int* __restrict__ rc_ctx, const int* __restrict__ rc_inter,
    const int* __restrict__ rc_half,
    int* __restrict__ ro_ctx, int* __restrict__ ro_inter, int* __restrict__ ro_half,
    int* __restrict__ header)
{
  __shared__ int lds[1024];
  const int tid = threadIdx.x;
  const int per = NTOT / 1024;                      // 32 rows per thread
  const int* rcs[3] = { rc_ctx, rc_inter, rc_half };
  int*       ros[3] = { ro_ctx, ro_inter, ro_half };
  int totals[3];

  for (int arr = 0; arr < 3; ++arr) {
    const int* rc = rcs[arr];
    int*       ro = ros[arr];
    int local[32];
    int s = 0;
    for (int k = 0; k < per; ++k) { local[k] = rc[tid * per + k]; s += local[k]; }

    lds[tid] = s;
    __syncthreads();
    for (int off = 1; off < 1024; off <<= 1) {      // Hillis-Steele inclusive
      int x = (tid >= off) ? lds[tid - off] : 0;
      __syncthreads();
      lds[tid] += x;
      __syncthreads();
    }
    int run = lds[tid] - s;                         // exclusive prefix
    totals[arr] = lds[1023];
    __syncthreads();
    for (int k = 0; k < per; ++k) { ro[tid * per + k] = run; run += local[k]; }
  }

  if (tid == 0) {
    header[0] = totals[0];                                           // radial ctx
    header[1] = totals[1];                                           // inter
    header[2] = totals[2];                                           // half
    header[3] = totals[0] + NB * GN_PER_GRAPH + NB * GG_PER_GRAPH;   // total ctx
  }
}

// ---------------------------------------------------------------------------
// Pass 4: closed-form global<->segment and global<->global edges, appended
// after the radial ctx edges in row-major order.
// ---------------------------------------------------------------------------
__global__ __launch_bounds__(512) void global_edges_kernel(
    const int* __restrict__ header, int* __restrict__ out)
{
  const int g    = blockIdx.x;
  const int l    = threadIdx.x;
  const int Ectx = header[3];
  const int rad  = header[0];
  const int base = g * NPG;
  const int gb   = rad + g * GN_PER_GRAPH;
  const int ggb  = rad + NB * GN_PER_GRAPH;

  if (l == 0) {                                    // row = ligand global node
    for (int k = 0; k < LIG - 1; ++k) {
      int p = gb + k;
      out[p] = base; out[Ectx + p] = base + 1 + k;
    }
    int p = ggb + g * 2;                           // (g0, g1)
    out[p] = base; out[Ectx + p] = base + LIG;
  } else if (l < LIG) {                            // normal ligand rows
    int p = gb + (LIG - 1) + (l - 1);
    out[p] = base + l; out[Ectx + p] = base;
  } else if (l == LIG) {                           // row = protein global node
    for (int k = 0; k < NPG - LIG - 1; ++k) {
      int p = gb + 2 * (LIG - 1) + k;
      out[p] = base + LIG; out[Ectx + p] = base + LIG + 1 + k;
    }
    int p = ggb + g * 2 + 1;                       // (g1, g0)
    out[p] = base + LIG; out[Ectx + p] = base;
  } else {                                         // normal protein rows
    int p = gb + 2 * (LIG - 1) + (NPG - LIG - 1) + (l - LIG - 1);
    out[p] = base + l; out[Ectx + p] = base + LIG;
  }
}

// ---------------------------------------------------------------------------
extern "C" void kernel_launch(void* const* d_in, const int* in_sizes, int n_in,
                              void* d_out, int out_size, void* d_ws, size_t ws_size,
                              hipStream_t stream) {
  (void)in_sizes; (void)n_in; (void)out_size; (void)ws_size;
  const float* X = (const float*)d_in[0];
  int* out = (int*)d_out;

  char* ws = (char*)d_ws;
  size_t o = 0;
  float4* xs4   = (float4*)(ws + o); o += (size_t)NTOT * sizeof(float4);
  int* rc_ctx   = (int*)(ws + o);    o += (size_t)NTOT * sizeof(int);
  int* rc_inter = (int*)(ws + o);    o += (size_t)NTOT * sizeof(int);
  int* rc_half  = (int*)(ws + o);    o += (size_t)NTOT * sizeof(int);
  int* ro_ctx   = (int*)(ws + o);    o += (size_t)NTOT * sizeof(int);
  int* ro_inter = (int*)(ws + o);    o += (size_t)NTOT * sizeof(int);
  int* ro_half  = (int*)(ws + o);    o += (size_t)NTOT * sizeof(int);
  int* header   = (int*)(ws + o);

  prep_kernel<<<(NTOT + 255) / 256, 256, 0, stream>>>(X, xs4);
  tiles_kernel<0><<<NB * TPG, 32, 0, stream>>>(xs4,
      rc_ctx, rc_inter, rc_half, ro_ctx, ro_inter, ro_half, header, out);
  scan_kernel<<<1, 1024, 0, stream>>>(rc_ctx, rc_inter, rc_half,
      ro_ctx, ro_inter, ro_half, header);
  tiles_kernel<1><<<NB * TPG, 32, 0, stream>>>(xs4,
      rc_ctx, rc_inter, rc_half, ro_ctx, ro_inter, ro_half, header, out);
  global_edges_kernel<<<NB, 512, 0, stream>>>(header, out);
}